// ParametricRotationLayer_35931696398700
// MI455X (gfx1250) — compile-verified
//
#include <hip/hip_runtime.h>
#include <math.h>

// ---------------------------------------------------------------------------
// ParametricRotationLayer: out[b,n,k] = sum_j states[b,n,j] * U(n)[j,k]
// U(n) = Rz(tz)Ry(ty)Rx(tx) in SU(2):  U = [[a, b], [-conj(b), conj(a)]]
//
// Roofline: 32 B HBM traffic per element (16 B in, 16 B out) x 16.7M elements
// = 536 MB -> ~23 us at 23.3 TB/s; ~16 FMAs/element => hard bandwidth-bound.
// WMMA cannot help (2x2 per-qubit matrices, different per n; mapping to
// 16x16x4 f32 WMMA gives 25% utilization and strided B-axis loads).
// CDNA5 paths used: async global->LDS staging (ASYNCcnt) for the U table,
// non-temporal TH on the 536 MB streams so L2 keeps U hot.
// ---------------------------------------------------------------------------

typedef float f4  __attribute__((ext_vector_type(4)));
typedef int   v4i __attribute__((vector_size(16)));   // matches builtin's V4i

#define NQ      4096   // qubits
#define NB      4096   // batch
#define NPT     2      // qubits per thread (-> b128 state loads)
#define THREADS 256
#define TILE_N  (THREADS * NPT)   // 512 qubits per block
#define ROWS    16                // batch rows per block

// ---------------- Kernel 1: per-qubit SU(2) parameters (alpha, beta) -------
__global__ __launch_bounds__(256) void qrot_build_u(
    const float* __restrict__ tx, const float* __restrict__ ty,
    const float* __restrict__ tz, f4* __restrict__ U) {
  int n = blockIdx.x * blockDim.x + threadIdx.x;
  if (n >= NQ) return;
  float hx = 0.5f * tx[n], hy = 0.5f * ty[n], hz = 0.5f * tz[n];
  float cx = cosf(hx), sx = sinf(hx);
  float cy = cosf(hy), sy = sinf(hy);
  float cz = cosf(hz), sz = sinf(hz);
  f4 u;
  u.x =  cz * cy * cx + sz * sy * sx;   // alpha.re  (U00.re)
  u.y =  cz * sy * sx - sz * cy * cx;   // alpha.im  (U00.im)
  u.z = -cz * sy * cx - sz * cy * sx;   // beta.re   (U01.re)
  u.w =  sz * sy * cx - cz * cy * sx;   // beta.im   (U01.im)
  U[n] = u;
}

// ---------------- async global->LDS staging helpers ------------------------
__device__ __forceinline__ void async_stage_b128(const void* gsrc, void* lds_dst) {
#if __has_builtin(__builtin_amdgcn_global_load_async_to_lds_b128)
  // Signature (probe-confirmed): (v4i AS1* src, v4i AS3* dst, imm off, imm cpol)
  __builtin_amdgcn_global_load_async_to_lds_b128(
      (__attribute__((address_space(1))) v4i*)(unsigned long long)gsrc,
      (__attribute__((address_space(3))) v4i*)(unsigned)(unsigned long long)lds_dst,
      0, 0);
#else
  unsigned lds_addr = (unsigned)(unsigned long long)lds_dst;
  asm volatile("global_load_async_to_lds_b128 %0, %1, off"
               :: "v"(lds_addr), "v"(gsrc) : "memory");
#endif
}

__device__ __forceinline__ void async_wait0() {
#if __has_builtin(__builtin_amdgcn_s_wait_asynccnt)
  __builtin_amdgcn_s_wait_asynccnt(0);
#else
  asm volatile("s_wait_asynccnt 0" ::: "memory");
#endif
}

// ---------------- complex 2-vector x SU(2) matrix --------------------------
__device__ __forceinline__ f4 apply_su2(float a0, float a1, float b0, float b1,
                                        f4 u) {
  // s0=(a0+ib0), s1=(a1+ib1); alpha=(u.x,u.y), beta=(u.z,u.w)
  // out0 = s0*alpha - s1*conj(beta); out1 = s0*beta + s1*conj(alpha)
  f4 o;
  o.x = a0 * u.x - b0 * u.y - a1 * u.z - b1 * u.w;   // out0.re
  o.y = a0 * u.y + b0 * u.x + a1 * u.w - b1 * u.z;   // out0.im
  o.z = a0 * u.z - b0 * u.w + a1 * u.x + b1 * u.y;   // out1.re
  o.w = a0 * u.w + b0 * u.z - a1 * u.y + b1 * u.x;   // out1.im
  return o;
}

// ---------------- Kernel 2: apply U to the state stream --------------------
__global__ __launch_bounds__(THREADS) void qrot_apply(
    const float* __restrict__ sre, const float* __restrict__ sim,
    const f4* __restrict__ U, float* __restrict__ out) {
  __shared__ f4 Ulds[TILE_N];   // 8 KB

  const int t  = threadIdx.x;
  const int n0 = blockIdx.x * TILE_N + NPT * t;  // first of 2 owned qubits
  const int b0 = blockIdx.y * ROWS;              // first batch row of the tile

  // Stage this block's 512 qubit unitaries (8 KB) into LDS via the CDNA5
  // async global->LDS path (tracked on ASYNCcnt). Each lane consumes only
  // entries it staged itself, so a wave-local s_wait_asynccnt suffices.
  async_stage_b128((const void*)(U + n0),     (void*)&Ulds[NPT * t]);
  async_stage_b128((const void*)(U + n0 + 1), (void*)&Ulds[NPT * t + 1]);
  async_wait0();

  const f4 u0 = Ulds[NPT * t];
  const f4 u1 = Ulds[NPT * t + 1];

  const f4* __restrict__ pr = (const f4*)sre;  // 2 qubits' (re0,re1) per f4
  const f4* __restrict__ pi = (const f4*)sim;
  f4*       __restrict__ po = (f4*)out;        // one qubit's (c0,c1) per f4

#pragma unroll 2
  for (int r = 0; r < ROWS; ++r) {
    const int idx = (b0 + r) * NQ + n0;        // even; <= 2^24, fits int
    // Streaming data: non-temporal so the 192 MB L2 keeps the U table hot.
    const f4 a = __builtin_nontemporal_load(pr + (idx >> 1)); // a0,a1,a0',a1'
    const f4 b = __builtin_nontemporal_load(pi + (idx >> 1)); // b0,b1,b0',b1'
    const f4 o0 = apply_su2(a.x, a.y, b.x, b.y, u0);  // qubit n0
    const f4 o1 = apply_su2(a.z, a.w, b.z, b.w, u1);  // qubit n0+1
    __builtin_nontemporal_store(o0, po + idx);
    __builtin_nontemporal_store(o1, po + idx + 1);
  }
}

// ---------------------------------------------------------------------------
extern "C" void kernel_launch(void* const* d_in, const int* in_sizes, int n_in,
                              void* d_out, int out_size, void* d_ws, size_t ws_size,
                              hipStream_t stream) {
  const float* sre = (const float*)d_in[0];   // (B,N,2) f32
  const float* sim = (const float*)d_in[1];   // (B,N,2) f32
  const float* tx  = (const float*)d_in[2];   // (N,) f32
  const float* ty  = (const float*)d_in[3];
  const float* tz  = (const float*)d_in[4];
  float* out = (float*)d_out;                 // (B,N,2) complex64 -> interleaved f32
  f4* U = (f4*)d_ws;                          // 4096 * 16 B = 64 KB scratch

  qrot_build_u<<<NQ / 256, 256, 0, stream>>>(tx, ty, tz, U);

  dim3 grid(NQ / TILE_N, NB / ROWS);          // 8 x 256 = 2048 blocks
  qrot_apply<<<grid, THREADS, 0, stream>>>(sre, sim, U, out);
}